// TvarLayer_14577119003346
// MI455X (gfx1250) — compile-verified
//
#include <hip/hip_runtime.h>
#include <cstdint>

// Problem constants (fixed by the reference's setup_inputs)
constexpr int Bn = 4, C = 8, H = 128, Wd = 128;   // image
constexpr int Nf = 8, K = 9;                      // filters, 3x3 taps
constexpr int L  = H * Wd;                        // 16384 pixels/image
constexpr int TILE_H = 8;                         // rows per workgroup
constexpr int LDS_W  = 136;                       // padded row stride (floats), 544B = 16B aligned
// LDS column layout: global col gc in [-1, 128] stored at lc = gc + 4  (lc in 3..132)
// -> interior (gc=0..127) starts at lc=4: 16B-aligned dest for b128 async DMA

typedef __attribute__((ext_vector_type(4))) float v4f;

__global__ __launch_bounds__(256) void tvar_kernel(
    const float* __restrict__ arr,   // [B, C, H, W]
    const float* __restrict__ Wp,    // [N, C, K, 1]
    const float* __restrict__ Bp,    // [N, C, 1, 1]
    float* __restrict__ out)         // [B, N*C*K, L]
{
    __shared__ float tile[(TILE_H + 2) * LDS_W];  // 10*136*4 = 5440 B

    const int blk  = blockIdx.x;
    const int tidx = blk % (H / TILE_H);            // 0..15 row tile
    const int c    = (blk / (H / TILE_H)) % C;      // channel
    const int b    = blk / ((H / TILE_H) * C);      // batch
    const int row0 = tidx * TILE_H;

    const float* img = arr + (size_t)(b * C + c) * (H * Wd);

    // ---- Zero the always-OOB halo columns (gc = -1 and gc = 128) ----------
    if (threadIdx.x < 20) {
        const int r    = threadIdx.x >> 1;
        const int side = threadIdx.x & 1;
        tile[r * LDS_W + (side ? 132 : 3)] = 0.0f;
    }

    // ---- Stage 10 rows x 128 interior cols via async b128 DMA -------------
    // 10 rows * 32 chunks of 4 floats = 320 transfers; OOB rows -> zero b128.
    for (int i = threadIdx.x; i < 10 * 32; i += 256) {
        const int r  = i >> 5;         // 0..9  (global row = row0-1+r)
        const int ch = i & 31;         // 0..31 (global col = 4*ch)
        const int gr = row0 - 1 + r;
        float* lp = &tile[r * LDS_W + 4 + ch * 4];      // 16B aligned
        if ((gr >= 0) & (gr < H)) {
            uint32_t lds_off = (uint32_t)(uintptr_t)lp; // low 32 bits = LDS offset
            uint64_t gaddr   = (uint64_t)(uintptr_t)(img + gr * Wd + ch * 4);
            asm volatile("global_load_async_to_lds_b128 %0, %1, off"
                         :
                         : "v"(lds_off), "v"(gaddr)
                         : "memory");
        } else {
            *(v4f*)lp = (v4f)0.0f;     // zero padding rows (first/last tile)
        }
    }
#if __has_builtin(__builtin_amdgcn_s_wait_asynccnt)
    __builtin_amdgcn_s_wait_asynccnt(0);
#else
    asm volatile("s_wait_asynccnt 0" ::: "memory");
#endif
    __syncthreads();

    // ---- Per-thread pixel block: 1 row x 4 consecutive columns ------------
    const int lane = threadIdx.x & 31;   // 0..31 -> cols 4*lane .. 4*lane+3
    const int row  = threadIdx.x >> 5;   // 0..7 tile row
    const int col4 = lane * 4;

    // 3x6 patch footprint covering all four 3x3 windows -> registers
    float p[3][6];
#pragma unroll
    for (int dr = 0; dr < 3; ++dr)
#pragma unroll
        for (int j = 0; j < 6; ++j)
            p[dr][j] = tile[(row + dr) * LDS_W + (col4 + 3 + j)];

    const int l0 = (row0 + row) * Wd + col4;   // linear pixel index of first col

#pragma unroll
    for (int n = 0; n < Nf; ++n) {
        // uniform (workgroup-scalar) weight fetch
        const float* wp = Wp + (size_t)((n * C + c) * K);
        float w[9];
        float s = 0.0f;
#pragma unroll
        for (int k = 0; k < 9; ++k) { w[k] = wp[k]; s += w[k]; }
        const float inv  = 1.0f / s;               // divide by size = sum_k w
        const float bias = Bp[n * C + c];

        // per-pixel mean over the 9 taps
        float mean[4];
#pragma unroll
        for (int j0 = 0; j0 < 4; ++j0) {
            float sum = 0.0f;
#pragma unroll
            for (int k = 0; k < 9; ++k)
                sum += p[k / 3][j0 + (k % 3)] * w[k];
            mean[j0] = sum * (1.0f / 9.0f);
        }

        float* ob = out + (size_t)((b * Nf + n) * C + c) * (size_t)(K * L) + l0;
#pragma unroll
        for (int k = 0; k < 9; ++k) {
            v4f v;
#pragma unroll
            for (int j0 = 0; j0 < 4; ++j0) {
                const float m = p[k / 3][j0 + (k % 3)] * w[k];
                const float d = m - mean[j0];
                v[j0] = fmaf(d * d, inv, bias);
            }
            // write-once 151MB output: NT b128, 512B contiguous per wave-store,
            // k*L folds into the 24-bit immediate offset (k * 64KB)
            __builtin_nontemporal_store(v, (v4f*)(ob + (size_t)k * L));
        }
    }
}

extern "C" void kernel_launch(void* const* d_in, const int* in_sizes, int n_in,
                              void* d_out, int out_size, void* d_ws, size_t ws_size,
                              hipStream_t stream) {
    const float* arr = (const float*)d_in[0];   // array [4,8,128,128] f32
    const float* Wp  = (const float*)d_in[1];   // W     [8,8,9,1]    f32
    const float* Bp  = (const float*)d_in[2];   // Bbias [8,8,1,1]    f32
    float* out = (float*)d_out;                 // [4,576,16384]      f32

    const int grid = Bn * C * (H / TILE_H);     // 512 workgroups
    tvar_kernel<<<grid, 256, 0, stream>>>(arr, Wp, Bp, out);
}